// AutoSelectAttention_73976516706635
// MI455X (gfx1250) — compile-verified
//
#include <hip/hip_runtime.h>

// y[bh,m,k] = -((x_k + mean)/(soft+EPS))^2 + intercept,  x_k = k - (L-1)
// Factored as rank-3 GEMM:  y[r,k] = a0*x^2 + a1*x + a2
//   a0 = -inv^2, a1 = -2*mean*inv^2, a2 = intercept - mean^2*inv^2, inv = 1/(soft+EPS)
// -> V_WMMA_F32_16X16X4_F32 tiles (K=3 padded to 4 with zeros).

typedef __attribute__((ext_vector_type(2))) float v2f;
typedef __attribute__((ext_vector_type(8))) float v8f;

#define L_CONST   1024
#define NROWS     (32 * 1024)          // BH * M = 32768 rows
#define NCOLS     (2 * L_CONST - 1)    // 2047 columns
#define TILES_N   128                  // ceil(2047/16) -> covers 2048, last col masked
#define TILES_PER_WAVE (TILES_N / 8)   // 8 waves per block (256 threads, wave32)
#define EPS_CONST 1e-5f

__global__ __launch_bounds__(256)
void span_parabola_wmma_kernel(const float* __restrict__ span,
                               float* __restrict__ out) {
    const int lane = threadIdx.x & 31;   // wave32
    const int wave = threadIdx.x >> 5;   // 0..7
    const int r0   = blockIdx.x * 16;    // row tile base (gridDim.x == NROWS/16)

    // ---- Build A fragment (16x4 fp32, 2 VGPRs per lane) -------------------
    // Lanes 0-15:  VGPR0 = A[m][K=0] = a0,  VGPR1 = A[m][K=1] = a1
    // Lanes 16-31: VGPR0 = A[m][K=2] = a2,  VGPR1 = A[m][K=3] = 0
    const int   mrow  = r0 + (lane & 15);
    const float mean  = span[mrow * 3 + 0];
    const float soft  = span[mrow * 3 + 1];
    const float icept = span[mrow * 3 + 2];

    const float inv = 1.0f / (soft + EPS_CONST);
    const float n2  = inv * inv;
    const float a0  = -n2;
    const float a1  = -2.0f * mean * n2;
    const float a2  = icept - mean * mean * n2;

    v2f A;
    A.x = (lane < 16) ? a0 : a2;
    A.y = (lane < 16) ? a1 : 0.0f;

    // Each wave sweeps a strip of 16 column tiles, reusing A.
    const int tn_base = wave * TILES_PER_WAVE;

    for (int t = 0; t < TILES_PER_WAVE; ++t) {
        const int c0  = (tn_base + t) * 16;     // column tile base
        const int col = c0 + (lane & 15);       // N = lane % 16
        const float x = (float)(col - (L_CONST - 1));

        // ---- B fragment (4x16 fp32, 2 VGPRs per lane) ---------------------
        // Lanes 0-15:  VGPR0 = B[K=0][n] = x^2, VGPR1 = B[K=1][n] = x
        // Lanes 16-31: VGPR0 = B[K=2][n] = 1,   VGPR1 = B[K=3][n] = 0
        v2f B;
        B.x = (lane < 16) ? (x * x) : 1.0f;
        B.y = (lane < 16) ? x       : 0.0f;

        v8f C = {};
        // 8 args: (neg_a, A, neg_b, B, c_mod, C, reuse_a, reuse_b)
        C = __builtin_amdgcn_wmma_f32_16x16x4_f32(
                false, A, false, B, (short)0, C, false, false);

        // ---- Store 16x16 tile --------------------------------------------
        // D VGPR r, lane L -> element (M = r + 8*(L>=16), N = L%16)
        const int rbase = r0 + ((lane >> 4) << 3);
        if (col < NCOLS) {   // masks only tile_n==127, n==15 (stores only; WMMA done)
#pragma unroll
            for (int r = 0; r < 8; ++r) {
                const size_t idx = (size_t)(rbase + r) * (size_t)NCOLS + (size_t)col;
                __builtin_nontemporal_store(C[r], &out[idx]);  // streaming: output > L2
            }
        }
    }
}

extern "C" void kernel_launch(void* const* d_in, const int* in_sizes, int n_in,
                              void* d_out, int out_size, void* d_ws, size_t ws_size,
                              hipStream_t stream) {
    (void)in_sizes; (void)n_in; (void)d_ws; (void)ws_size; (void)out_size;
    const float* span = (const float*)d_in[0];   // (32, 1024, 3) fp32
    float* out = (float*)d_out;                  // (32, 1024, 2047) fp32

    // One block (8 waves) per 16-row tile; each wave sweeps 16 column tiles.
    const int grid = NROWS / 16;                 // 2048 blocks
    span_parabola_wmma_kernel<<<grid, 256, 0, stream>>>(span, out);
}